// Net_25804163514984
// MI455X (gfx1250) — compile-verified
//
#include <hip/hip_runtime.h>
#include <math.h>

// ---------------------------------------------------------------------------
// GAT pipeline for CDNA5 / gfx1250 (wave32).
//   - Dense feature transforms via V_WMMA_F32_16X16X4_F32 (full fp32; the
//     workload is HBM-bound so fp32 WMMA is the optimal precision point).
//   - B operand staged zero-padded in LDS -> unconditional ds_loads in the
//     unrolled WMMA loop (no divergent exec-masked B loads).
//   - Edge phase uses L2-resident atomics (int-ordered atomicMax for segment
//     max, atomicAdd for softmax-sum + weighted aggregation).
// ---------------------------------------------------------------------------

typedef __attribute__((ext_vector_type(2))) float v2f;
typedef __attribute__((ext_vector_type(8))) float v8f;

#ifndef __has_builtin
#define __has_builtin(x) 0
#endif

#if __has_builtin(__builtin_amdgcn_wmma_f32_16x16x4_f32)
#define HAVE_WMMA_F32_16X16X4 1
#else
#define HAVE_WMMA_F32_16X16X4 0
#endif

__device__ __forceinline__ float lrelu(float v, float s) { return v > 0.f ? v : v * s; }
// Monotone float->int map so signed-int atomicMax implements float max.
__device__ __forceinline__ int enc_ord(float f) {
    int i = __float_as_int(f);
    return i >= 0 ? i : (i ^ 0x7fffffff);
}
__device__ __forceinline__ float dec_ord(int i) {
    return __int_as_float(i >= 0 ? i : (i ^ 0x7fffffff));
}

// ---------------------------------------------------------------------------
// GEMM: C[M,Nout] = act(A[M,K] @ B[K,Nout] + bias), Nout <= 16, K % 4 == 0.
// One wave handles a 16-row tile with V_WMMA_F32_16X16X4_F32 (K unrolled).
// A-frag (16x4 f32): lane L -> M = L%16; VGPR0 = K+2*(L/16), VGPR1 = +1.
// B-frag (4x16 f32): lane L -> N = L%16; VGPR0 = K+2*(L/16), VGPR1 = +1.
// C/D (16x16 f32, 8 VGPRs): vgpr r, lane L -> (M = r + 8*(L/16), N = L%16).
// ---------------------------------------------------------------------------
#if HAVE_WMMA_F32_16X16X4
template <int K>
__global__ void gemm_wmma(const float* __restrict__ A, int lda,
                          const float* __restrict__ B, int ldb,
                          float* __restrict__ C, int ldc,
                          int M, int Nout,
                          const float* __restrict__ bias, float slope) {
    __shared__ float ldsB[K * 16];
    // Cooperative zero-padded fill of B into LDS (once per block).
    for (int idx = threadIdx.x; idx < K * 16; idx += blockDim.x) {
        const int k = idx >> 4, n = idx & 15;
        ldsB[idx] = (n < Nout) ? B[(size_t)k * ldb + n] : 0.f;
    }
    __syncthreads();

    const int wave = blockIdx.x * (blockDim.x >> 5) + (threadIdx.x >> 5);
    const int lane = threadIdx.x & 31;
    const int row0 = wave << 4;
    if (row0 >= M) return;                 // wave-uniform: EXEC stays all-ones
    const int mrow = lane & 15;
    const int hi   = lane >> 4;            // 0 or 1
    const int kh   = hi << 1;              // 0 or 2
    const int ncol = lane & 15;
    int arowi = row0 + mrow;
    if (arowi >= M) arowi = M - 1;         // safe for partial tiles
    const float* arow = A + (size_t)arowi * lda;

    v8f acc = {};
#pragma unroll
    for (int k0 = 0; k0 < K; k0 += 4) {
        v2f a, b;
        a.x = arow[k0 + kh + 0];
        a.y = arow[k0 + kh + 1];
        b.x = ldsB[(k0 + kh + 0) * 16 + ncol];
        b.y = ldsB[(k0 + kh + 1) * 16 + ncol];
        acc = __builtin_amdgcn_wmma_f32_16x16x4_f32(
            /*neg_a=*/false, a, /*neg_b=*/false, b,
            /*c_mod=*/(short)0, acc, /*reuse_a=*/false, /*reuse_b=*/false);
    }
    if (ncol < Nout) {
        const float bv = bias ? bias[ncol] : 0.f;
#pragma unroll
        for (int r = 0; r < 8; ++r) {
            const int row = row0 + r + (hi << 3);
            if (row < M) {
                float v = acc[r] + bv;
                C[(size_t)row * ldc + ncol] = lrelu(v, slope); // slope==1 -> id
            }
        }
    }
}
#endif

// Fallback (launched if the f32 WMMA builtin is unavailable or K is not 128, 16 or 8).
__global__ void gemm_scalar(const float* __restrict__ A, int lda,
                            const float* __restrict__ B, int ldb,
                            float* __restrict__ C, int ldc,
                            int M, int K, int Nout,
                            const float* __restrict__ bias, float slope) {
    int idx = blockIdx.x * blockDim.x + threadIdx.x;
    int row = idx / Nout, col = idx - row * Nout;
    if (row >= M) return;
    float s = bias ? bias[col] : 0.f;
    for (int k = 0; k < K; ++k) s += A[(size_t)row * lda + k] * B[(size_t)k * ldb + col];
    C[(size_t)row * ldc + col] = lrelu(s, slope);
}

// ---------------------------------------------------------------------------
// Node / edge kernels (templated on <H, C> so gathers vectorize to b128 and
// atomic loops fully unroll).
// ---------------------------------------------------------------------------

// alpha_src/alpha_dst dot products; init segment-max with self-loop logit.
template <int H, int C>
__global__ void node_alpha_init(const float* __restrict__ xl,
                                const float* __restrict__ av_src,
                                const float* __restrict__ av_dst,
                                float* __restrict__ asrc, float* __restrict__ adst,
                                int* __restrict__ mkey, int N) {
    int i = blockIdx.x * blockDim.x + threadIdx.x;
    if (i >= N) return;
    const float* row = xl + (size_t)i * (H * C);
#pragma unroll
    for (int h = 0; h < H; ++h) {
        float ss = 0.f, dd = 0.f;
#pragma unroll
        for (int c = 0; c < C; ++c) {
            float v = row[h * C + c];
            ss += v * av_src[h * C + c];
            dd += v * av_dst[h * C + c];
        }
        asrc[(size_t)i * H + h] = ss;
        adst[(size_t)i * H + h] = dd;
        mkey[(size_t)i * H + h] = enc_ord(lrelu(ss + dd, 0.2f));  // self-loop edge
    }
}

// Pass 1 over edges: segment max via int-encoded atomicMax (L2-resident).
template <int H>
__global__ void edge_max_k(const int* __restrict__ src, const int* __restrict__ dst,
                           const float* __restrict__ asrc, const float* __restrict__ adst,
                           int* mkey, int E) {
    int e = blockIdx.x * blockDim.x + threadIdx.x;
    if (e >= E) return;
    int s = src[e], d = dst[e];
#pragma unroll
    for (int h = 0; h < H; ++h) {
        float v = lrelu(asrc[(size_t)s * H + h] + adst[(size_t)d * H + h], 0.2f);
        atomicMax(&mkey[(size_t)d * H + h], enc_ord(v));
    }
}

// Decode max; seed softmax-sum and accumulator with the self-loop contribution.
// (mkey and mval alias the same buffer: read-then-overwrite within a thread.)
template <int H, int C>
__global__ void node_init_acc(const float* __restrict__ xl,
                              const float* __restrict__ asrc,
                              const float* __restrict__ adst,
                              int* mkey, float* mval,
                              float* __restrict__ sbuf, float* __restrict__ acc,
                              int N) {
    int i = blockIdx.x * blockDim.x + threadIdx.x;
    if (i >= N) return;
#pragma unroll
    for (int h = 0; h < H; ++h) {
        float m = dec_ord(mkey[(size_t)i * H + h]);
        mval[(size_t)i * H + h] = m;
        float e = lrelu(asrc[(size_t)i * H + h] + adst[(size_t)i * H + h], 0.2f);
        float w = __expf(e - m);
        sbuf[(size_t)i * H + h] = w;
#pragma unroll
        for (int c = 0; c < C; ++c)
            acc[(size_t)i * H * C + h * C + c] = w * xl[(size_t)i * (H * C) + h * C + c];
    }
}

// Pass 2 over edges: fused softmax-sum + unnormalized weighted aggregation.
template <int H, int C>
__global__ void edge_acc_k(const int* __restrict__ src, const int* __restrict__ dst,
                           const float* __restrict__ asrc, const float* __restrict__ adst,
                           const float* __restrict__ mval,
                           const float* __restrict__ xl,
                           float* sbuf, float* acc, int E) {
    int e = blockIdx.x * blockDim.x + threadIdx.x;
    if (e >= E) return;
    int s = src[e], d = dst[e];
#pragma unroll
    for (int h = 0; h < H; ++h) {
        float v = lrelu(asrc[(size_t)s * H + h] + adst[(size_t)d * H + h], 0.2f);
        float w = __expf(v - mval[(size_t)d * H + h]);
        atomicAdd(&sbuf[(size_t)d * H + h], w);
        const float* xr = xl + (size_t)s * (H * C) + h * C;
        float* ar = acc + (size_t)d * H * C + h * C;
#pragma unroll
        for (int c = 0; c < C; ++c) atomicAdd(&ar[c], w * xr[c]);
    }
}

// Normalize, add bias, leaky-relu -> next layer input (concat layers).
template <int H, int C>
__global__ void node_finish(const float* __restrict__ acc, const float* __restrict__ sbuf,
                            const float* __restrict__ bias, float* __restrict__ hout,
                            int N, float slope) {
    int i = blockIdx.x * blockDim.x + threadIdx.x;
    if (i >= N) return;
#pragma unroll
    for (int h = 0; h < H; ++h) {
        float sinv = 1.f / sbuf[(size_t)i * H + h];
#pragma unroll
        for (int c = 0; c < C; ++c) {
            float v = acc[(size_t)i * H * C + h * C + c] * sinv + bias[h * C + c];
            hout[(size_t)i * (H * C) + h * C + c] = lrelu(v, slope);
        }
    }
}

// Final layer (H=1): normalize, add bias, log_softmax over C classes.
template <int C>
__global__ void node_finish_lsm(const float* __restrict__ acc, const float* __restrict__ sbuf,
                                const float* __restrict__ bias, float* __restrict__ out,
                                int N) {
    int i = blockIdx.x * blockDim.x + threadIdx.x;
    if (i >= N) return;
    float sinv = 1.f / sbuf[i];
    float v[C];
    float mx = -INFINITY;
#pragma unroll
    for (int c = 0; c < C; ++c) {
        float t = acc[(size_t)i * C + c] * sinv + bias[c];
        v[c] = t;
        mx = fmaxf(mx, t);
    }
    float se = 0.f;
#pragma unroll
    for (int c = 0; c < C; ++c) se += __expf(v[c] - mx);
    float lse = mx + __logf(se);
#pragma unroll
    for (int c = 0; c < C; ++c) out[(size_t)i * C + c] = v[c] - lse;
}

// ---------------------------------------------------------------------------
// Host orchestration
// ---------------------------------------------------------------------------
static void launch_gemm(hipStream_t st, const float* A, int lda, const float* B, int ldb,
                        float* C, int ldc, int M, int K, int Nout,
                        const float* bias, float slope) {
#if HAVE_WMMA_F32_16X16X4
    const int waves   = (M + 15) >> 4;
    const int threads = waves * 32;
    const int blocks  = (threads + 255) / 256;
    if (K == 128) {
        gemm_wmma<128><<<blocks, 256, 0, st>>>(A, lda, B, ldb, C, ldc, M, Nout, bias, slope);
    } else if (K == 16) {
        gemm_wmma<16><<<blocks, 256, 0, st>>>(A, lda, B, ldb, C, ldc, M, Nout, bias, slope);
    } else if (K == 8) {
        gemm_wmma<8><<<blocks, 256, 0, st>>>(A, lda, B, ldb, C, ldc, M, Nout, bias, slope);
    } else {
        int tot = M * Nout;
        gemm_scalar<<<(tot + 255) / 256, 256, 0, st>>>(A, lda, B, ldb, C, ldc, M, K, Nout,
                                                       bias, slope);
    }
#else
    int tot = M * Nout;
    gemm_scalar<<<(tot + 255) / 256, 256, 0, st>>>(A, lda, B, ldb, C, ldc, M, K, Nout,
                                                   bias, slope);
#endif
}

extern "C" void kernel_launch(void* const* d_in, const int* in_sizes, int n_in,
                              void* d_out, int out_size, void* d_ws, size_t ws_size,
                              hipStream_t stream) {
    (void)n_in; (void)out_size; (void)ws_size;
    const float* x     = (const float*)d_in[0];
    const int*   eidx  = (const int*)  d_in[1];
    const float* lin_w = (const float*)d_in[2];
    const float* lin_b = (const float*)d_in[3];
    const float* w1    = (const float*)d_in[4];
    const float* a1s   = (const float*)d_in[5];
    const float* a1d   = (const float*)d_in[6];
    const float* b1    = (const float*)d_in[7];
    const float* w2    = (const float*)d_in[8];
    const float* a2s   = (const float*)d_in[9];
    const float* a2d   = (const float*)d_in[10];
    const float* b2    = (const float*)d_in[11];
    const float* w3    = (const float*)d_in[12];
    const float* a3s   = (const float*)d_in[13];
    const float* a3d   = (const float*)d_in[14];
    const float* b3    = (const float*)d_in[15];

    const int NFEAT = 128;
    const int N = in_sizes[0] / NFEAT;
    const int E = in_sizes[1] / 2;
    const int* src = eidx;
    const int* dst = eidx + E;
    float* out = (float*)d_out;

    // Workspace layout (<32 MB total; per-layer node state is L2-resident).
    char* ws = (char*)d_ws;
    size_t off = 0;
    auto alloc = [&](size_t nfl) -> float* {
        float* p = (float*)(ws + off);
        off += nfl * sizeof(float);
        return p;
    };
    float* hA   = alloc((size_t)N * 16);  // layer activations (ping)
    float* hB   = alloc((size_t)N * 16);  // layer activations (pong)
    float* xl   = alloc((size_t)N * 16);  // per-layer transformed features
    float* asrc = alloc((size_t)N * 2);
    float* adst = alloc((size_t)N * 2);
    float* mbuf = alloc((size_t)N * 2);   // int keys, then decoded float max
    float* sbuf = alloc((size_t)N * 2);   // softmax denominators
    float* acc  = alloc((size_t)N * 16);  // weighted-message accumulators

    const int TB = 256;
    const int gN = (N + TB - 1) / TB;
    const int gE = (E + TB - 1) / TB;

    // h0 = leaky_relu(x @ lin_w + lin_b, 0.01)  -> [N,8]
    launch_gemm(stream, x, NFEAT, lin_w, 8, hA, 8, N, NFEAT, 8, lin_b, 0.01f);

    // ---- conv1: GATConv(8 -> 8, heads=2, concat) ----
    launch_gemm(stream, hA, 8, w1, 16, xl, 16, N, 8, 16, nullptr, 1.0f);
    node_alpha_init<2, 8><<<gN, TB, 0, stream>>>(xl, a1s, a1d, asrc, adst, (int*)mbuf, N);
    edge_max_k<2><<<gE, TB, 0, stream>>>(src, dst, asrc, adst, (int*)mbuf, E);
    node_init_acc<2, 8><<<gN, TB, 0, stream>>>(xl, asrc, adst, (int*)mbuf, mbuf, sbuf, acc, N);
    edge_acc_k<2, 8><<<gE, TB, 0, stream>>>(src, dst, asrc, adst, mbuf, xl, sbuf, acc, E);
    node_finish<2, 8><<<gN, TB, 0, stream>>>(acc, sbuf, b1, hB, N, 0.01f);

    // ---- conv2: GATConv(16 -> 8, heads=2, concat) ----
    launch_gemm(stream, hB, 16, w2, 16, xl, 16, N, 16, 16, nullptr, 1.0f);
    node_alpha_init<2, 8><<<gN, TB, 0, stream>>>(xl, a2s, a2d, asrc, adst, (int*)mbuf, N);
    edge_max_k<2><<<gE, TB, 0, stream>>>(src, dst, asrc, adst, (int*)mbuf, E);
    node_init_acc<2, 8><<<gN, TB, 0, stream>>>(xl, asrc, adst, (int*)mbuf, mbuf, sbuf, acc, N);
    edge_acc_k<2, 8><<<gE, TB, 0, stream>>>(src, dst, asrc, adst, mbuf, xl, sbuf, acc, E);
    node_finish<2, 8><<<gN, TB, 0, stream>>>(acc, sbuf, b2, hA, N, 0.01f);

    // ---- conv3: GATConv(16 -> 10, heads=1, mean) + log_softmax ----
    launch_gemm(stream, hA, 16, w3, 10, xl, 10, N, 16, 10, nullptr, 1.0f);
    node_alpha_init<1, 10><<<gN, TB, 0, stream>>>(xl, a3s, a3d, asrc, adst, (int*)mbuf, N);
    edge_max_k<1><<<gE, TB, 0, stream>>>(src, dst, asrc, adst, (int*)mbuf, E);
    node_init_acc<1, 10><<<gN, TB, 0, stream>>>(xl, asrc, adst, (int*)mbuf, mbuf, sbuf, acc, N);
    edge_acc_k<1, 10><<<gE, TB, 0, stream>>>(src, dst, asrc, adst, mbuf, xl, sbuf, acc, E);
    node_finish_lsm<10><<<gN, TB, 0, stream>>>(acc, sbuf, b3, out, N);
}